// Smartpool_75909251989640
// MI455X (gfx1250) — compile-verified
//
#include <hip/hip_runtime.h>

// ---------------- problem constants (match reference) ----------------
#define Bc 4
#define Tc 2048
#define Cc 512
#define FIc 2
#define Zc 6                      // 3 * FI
#define LMAX (Bc*Tc + Bc*Zc)      // 8216
#define MMAX (Bc*Tc/FIc + Bc)     // 4100
#define BANDCHUNK 256             // band segments staged in LDS per pass

// ws layout (float index): [0..3]=padCount(int) [4]=validSum [5]=nlSum [6]=mean
// [7]=M(int) [8]=target [9]=totalT(int) [10..14]=S prefix(int) [16..19]=lenb(int)
// [20..23]=rm(int) [24..27]=sizes(int) ; arrays: nl@64, cs@64+LMAX, lennz(int)@64+2*LMAX

typedef __attribute__((ext_vector_type(2))) float v2f;
typedef __attribute__((ext_vector_type(8))) float v8f;

__device__ __forceinline__ float blockReduceSum256(float v) {
    __shared__ float red[256];
    int tid = threadIdx.x;
    red[tid] = v; __syncthreads();
    for (int off = 128; off > 0; off >>= 1) {
        if (tid < off) red[tid] += red[tid + off];
        __syncthreads();
    }
    return red[0];
}

__device__ __forceinline__ int firstGT(const float* cs, int L, float x) {
    int lo = 0, hi = L;
    while (lo < hi) { int mid = (lo + hi) >> 1; if (cs[mid] > x) hi = mid; else lo = mid + 1; }
    return lo;
}
__device__ __forceinline__ int firstGE(const float* cs, int L, float x) {
    int lo = 0, hi = L;
    while (lo < hi) { int mid = (lo + hi) >> 1; if (cs[mid] >= x) hi = mid; else lo = mid + 1; }
    return lo;
}

// ---------- K1: pad counts + valid-element sum ----------
__global__ __launch_bounds__(256) void k_stats(const float* __restrict__ feat,
                                               const unsigned char* __restrict__ mask,
                                               float* wsf) {
    int* wsi = (int*)wsf;
    int r = blockIdx.x;                      // row in [0, B*T)
    int b = r / Tc;
    if (mask[r]) {
        if (threadIdx.x == 0) atomicAdd(&wsi[b], 1);
        return;
    }
    const float* row = feat + (size_t)r * Cc;
    float s = 0.f;
    for (int c = threadIdx.x; c < Cc; c += 256) s += row[c];
    s = blockReduceSum256(s);
    if (threadIdx.x == 0) atomicAdd(&wsf[4], s);
}

// ---------- K2: finalize scalars ----------
__global__ void k_final(float* wsf) {
    int* wsi = (int*)wsf;
    int totalT = Bc * Tc;
    for (int b = 0; b < Bc; b++) totalT -= wsi[b];
    float mean   = wsf[4] / ((float)totalT * (float)Cc);
    float target = (float)totalT / 2.0f + (float)Bc;     // total_T/FACTOR + B
    wsf[6] = mean;
    wsi[7] = (int)roundf(target);
    wsf[8] = target;
    wsi[9] = totalT;
    int s = 0; wsi[10] = 0;
    for (int b = 0; b < Bc; b++) {
        int lenb = Tc - wsi[b];
        wsi[16 + b] = lenb;
        s += lenb + Zc;
        wsi[11 + b] = s;                                  // S[1..4], S[4] = L
    }
}

// ---------- K3: nl[k] = L1 diff per concatenated row ----------
__global__ __launch_bounds__(256) void k_nl(const float* __restrict__ feat, float* wsf) {
    const int* wsi = (const int*)wsf;
    float* nl = wsf + 64;
    int L = wsi[14];
    int k = blockIdx.x;
    if (k >= L) return;
    int b, pos;
    if      (k < wsi[11]) { b = 0; pos = k; }
    else if (k < wsi[12]) { b = 1; pos = k - wsi[11]; }
    else if (k < wsi[13]) { b = 2; pos = k - wsi[12]; }
    else                  { b = 3; pos = k - wsi[13]; }
    int lenb = wsi[16 + b];
    if (pos >= lenb) { if (threadIdx.x == 0) nl[k] = 0.f; return; }
    float mean = wsf[6];
    const float* cur  = feat + (size_t)(b * Tc + pos) * Cc;
    const float* prev = (pos > 0) ? (cur - Cc) : nullptr;
    float s = 0.f;
    for (int c = threadIdx.x; c < Cc; c += 256) {
        float p = prev ? prev[c] : mean;
        s += fabsf(cur[c] - p);
    }
    s = blockReduceSum256(s);
    if (threadIdx.x == 0) { nl[k] = s; atomicAdd(&wsf[5], s); }
}

// ---------- K4: single-block scan -> normalized cumsum cs ----------
__global__ __launch_bounds__(1024) void k_scan(float* wsf) {
    const int* wsi = (const int*)wsf;
    float* nl = wsf + 64;
    float* cs = wsf + 64 + LMAX;
    int L = wsi[14];
    float scale = wsf[8] / wsf[5];
    __shared__ float buf[1024];
    float carry = 0.f;
    for (int base = 0; base < L; base += 1024) {
        int idx = base + threadIdx.x;
        float x = (idx < L) ? nl[idx] : 0.f;
        buf[threadIdx.x] = x; __syncthreads();
        for (int off = 1; off < 1024; off <<= 1) {
            float v = (threadIdx.x >= (unsigned)off) ? buf[threadIdx.x - off] : 0.f;
            __syncthreads();
            buf[threadIdx.x] += v;
            __syncthreads();
        }
        if (idx < L) cs[idx] = (carry + buf[threadIdx.x]) * scale;
        carry += buf[1023];
        __syncthreads();
    }
}

// ---------- K5: nonzero-support length per output row ----------
__global__ __launch_bounds__(256) void k_lennz(float* wsf) {
    const int* wsi = (const int*)wsf;
    const float* cs = wsf + 64 + LMAX;
    int* lennz = (int*)(wsf + 64 + 2 * LMAX);
    int M = wsi[7], L = wsi[14];
    int j = blockIdx.x * blockDim.x + threadIdx.x;
    if (j >= M) return;
    float fj = (float)j;
    int lo0 = firstGT(cs, L, fj);
    int hi0 = min(firstGE(cs, L, fj + 1.f) + 1, L);
    int low = -1, high = -1;
    for (int i = lo0; i < hi0; i++) {
        float ci = cs[i], cim1 = (i > 0) ? cs[i - 1] : 0.f;
        if (fminf(ci, fj + 1.f) - fmaxf(cim1, fj) > 0.f) { low = i; break; }
    }
    for (int i = hi0 - 1; i >= lo0; i--) {
        float ci = cs[i], cim1 = (i > 0) ? cs[i - 1] : 0.f;
        if (fminf(ci, fj + 1.f) - fmaxf(cim1, fj) > 0.f) { high = i; break; }
    }
    lennz[j] = (low >= 0) ? (high - low + 1) : 0;
}

// ---------- K6: border argmax -> remove indices, sizes ----------
__global__ __launch_bounds__(128) void k_borders(float* wsf) {
    int* wsi = (int*)wsf;
    const int* lennz = (const int*)(wsf + 64 + 2 * LMAX);
    __shared__ int bv[128], bi[128];
    int M = wsi[7];
    int tc = 0;
    for (int b = 0; b < Bc - 1; b++) {
        int theor = (Tc - wsi[b]) / FIc + 1;
        tc += theor;
        int sl = (int)((float)theor * 0.05f);
        int lo = max(tc - sl, 0), hi = min(tc + sl, M);
        int bestv = -1, besti = lo;
        for (int j = lo + threadIdx.x; j < hi; j += 128) {
            int v = lennz[j];
            if (v > bestv) { bestv = v; besti = j; }
        }
        bv[threadIdx.x] = bestv; bi[threadIdx.x] = besti; __syncthreads();
        for (int off = 64; off > 0; off >>= 1) {
            if (threadIdx.x < (unsigned)off) {
                int ov = bv[threadIdx.x + off], oi = bi[threadIdx.x + off];
                if (ov > bv[threadIdx.x] || (ov == bv[threadIdx.x] && oi < bi[threadIdx.x])) {
                    bv[threadIdx.x] = ov; bi[threadIdx.x] = oi;
                }
            }
            __syncthreads();
        }
        if (threadIdx.x == 0) wsi[20 + b] = bi[0];
        __syncthreads();
    }
    if (threadIdx.x == 0) {
        wsi[23] = M - 1;
        int prev = 0;                                   // start[0] = 0
        for (int b = 0; b < Bc; b++) {
            int st = wsi[20 + b] - b;                   // start[b+1] = rm[b] - b
            wsi[24 + b] = st - prev;                    // sizes[b]
            prev = st;
        }
    }
}

// ---------- K7: band-sparse WMMA GEMM fused with removal scatter ----------
// LDS-staged band metadata: cs[i], cs[i-1], and feature-row index (-1 => zero row).
__global__ __launch_bounds__(256) void k_gemm(const float* __restrict__ feat,
                                              float* wsf,
                                              float* __restrict__ out, int newT) {
    const int* wsi = (const int*)wsf;
    const float* cs = wsf + 64 + LMAX;
    int M = wsi[7], L = wsi[14];
    int j0 = blockIdx.x * 16;
    if (j0 >= M) return;

    __shared__ float s_cs[BANDCHUNK];
    __shared__ float s_csm1[BANDCHUNK];
    __shared__ int   s_row[BANDCHUNK];

    int lane = threadIdx.x & 31;
    int wv   = threadIdx.x >> 5;            // 8 waves (wave32)
    int nbase = wv * 64;                    // 64 columns per wave (4 N-tiles of 16)

    int rm0 = wsi[20], rm1 = wsi[21], rm2 = wsi[22], rm3 = wsi[23];
    int S1 = wsi[11], S2 = wsi[12], S3 = wsi[13];
    int len0 = wsi[16], len1 = wsi[17], len2 = wsi[18], len3 = wsi[19];

    // band of segments overlapping bins [j0, j0+16)
    int iLo = firstGT(cs, L, (float)j0);
    int iHi = min(firstGE(cs, L, (float)(j0 + 16)) + 1, L);

    v8f acc0 = {}, acc1 = {}, acc2 = {}, acc3 = {};
    int khalf = (lane >> 4) << 1;           // K offset 0 (lanes 0-15) or 2 (lanes 16-31)
    int mrow  = lane & 15;                  // A fragment: M index
    int ncol  = lane & 15;                  // B fragment: N index
    float jm  = (float)(j0 + mrow);

    for (int base = iLo; base < iHi; base += BANDCHUNK) {
        int n = min(BANDCHUNK, iHi - base);
        int npad = (n + 3) & ~3;
        // ---- cooperative stage of band metadata into LDS ----
        {
            int tid = threadIdx.x;
            if (tid < npad) {
                float ci = 0.f, cim1 = 0.f;
                int row = -1;
                if (tid < n) {
                    int i = base + tid;
                    ci   = cs[i];
                    cim1 = (i > 0) ? cs[i - 1] : 0.f;
                    int bb, pos, lenbb;
                    if      (i < S1) { bb = 0; pos = i;      lenbb = len0; }
                    else if (i < S2) { bb = 1; pos = i - S1; lenbb = len1; }
                    else if (i < S3) { bb = 2; pos = i - S2; lenbb = len2; }
                    else             { bb = 3; pos = i - S3; lenbb = len3; }
                    if (pos < lenbb) row = bb * Tc + pos;
                }
                s_cs[tid] = ci; s_csm1[tid] = cim1; s_row[tid] = row;
            }
        }
        __syncthreads();

        // ---- branch-free WMMA loop over staged chunk ----
        for (int kk = 0; kk < n; kk += 4) {
            v2f a; v2f b0, b1, b2, b3;
#pragma unroll
            for (int v = 0; v < 2; v++) {
                int idx = kk + khalf + v;
                float ci   = s_cs[idx];
                float cim1 = s_csm1[idx];
                int   row  = s_row[idx];
                // interp weight: overlap of [cs[i-1],cs[i]] with bin [jm, jm+1]
                a[v] = fminf(fmaxf(ci - jm, 0.f), 1.f) - fminf(fmaxf(cim1 - jm, 0.f), 1.f);
                float x0 = 0.f, x1 = 0.f, x2 = 0.f, x3 = 0.f;
                if (row >= 0) {
                    const float* xrow = feat + (size_t)row * Cc;
                    x0 = xrow[nbase + ncol];
                    x1 = xrow[nbase + 16 + ncol];
                    x2 = xrow[nbase + 32 + ncol];
                    x3 = xrow[nbase + 48 + ncol];
                }
                b0[v] = x0; b1[v] = x1; b2[v] = x2; b3[v] = x3;
            }
            acc0 = __builtin_amdgcn_wmma_f32_16x16x4_f32(false, a, false, b0, (short)0, acc0, false, false);
            acc1 = __builtin_amdgcn_wmma_f32_16x16x4_f32(false, a, false, b1, (short)0, acc1, false, false);
            acc2 = __builtin_amdgcn_wmma_f32_16x16x4_f32(false, a, false, b2, (short)0, acc2, false, false);
            acc3 = __builtin_amdgcn_wmma_f32_16x16x4_f32(false, a, false, b3, (short)0, acc3, false, false);
        }
        __syncthreads();
    }

    // scatter: C/D layout -> lane l, VGPR r holds (M = 8*(l>=16)+r, N = l&15)
    int mbase = (lane >> 4) << 3;
#pragma unroll
    for (int r = 0; r < 8; r++) {
        int j = j0 + mbase + r;
        if (j >= M) continue;
        if (j == rm0 || j == rm1 || j == rm2 || j == rm3) continue;
        int bb = (j > rm0) + (j > rm1) + (j > rm2);
        int t  = j - ((bb == 0) ? 0 : (wsi[20 + bb - 1] + 1));
        if (t >= 0 && t < newT && t < wsi[24 + bb]) {
            float* po = out + ((size_t)bb * newT + t) * Cc + nbase + ncol;
            po[0]  = acc0[r];
            po[16] = acc1[r];
            po[32] = acc2[r];
            po[48] = acc3[r];
        }
    }
}

// ---------- K8: padding rows -> zero, write padding mask (as float) ----------
__global__ __launch_bounds__(128) void k_finalout(const float* wsf, float* out, int newT) {
    const int* wsi = (const int*)wsf;
    int rid = blockIdx.x;
    int b = rid / newT, t = rid % newT;
    int sz = wsi[24 + b];
    bool pad = (t >= sz);
    if (threadIdx.x == 0)
        out[(size_t)Bc * newT * Cc + (size_t)b * newT + t] = pad ? 1.f : 0.f;
    if (pad) {
        float* row = out + ((size_t)b * newT + t) * Cc;
        for (int c = threadIdx.x; c < Cc; c += 128) row[c] = 0.f;
    }
}

extern "C" void kernel_launch(void* const* d_in, const int* in_sizes, int n_in,
                              void* d_out, int out_size, void* d_ws, size_t ws_size,
                              hipStream_t stream) {
    const float* feat = (const float*)d_in[0];
    const unsigned char* mask = (const unsigned char*)d_in[1];
    float* out = (float*)d_out;
    float* wsf = (float*)d_ws;
    (void)in_sizes; (void)n_in; (void)ws_size;

    // out = [B,new_T,C] f32 ++ [B,new_T] mask  => out_size = B*new_T*(C+1)
    int newT = out_size / (Bc * (Cc + 1));

    hipMemsetAsync(d_ws, 0, 256, stream);                              // scalars/atomics
    k_stats   <<<Bc * Tc,           256,  0, stream>>>(feat, mask, wsf);
    k_final   <<<1,                 1,    0, stream>>>(wsf);
    k_nl      <<<LMAX,              256,  0, stream>>>(feat, wsf);
    k_scan    <<<1,                 1024, 0, stream>>>(wsf);
    k_lennz   <<<(MMAX + 255) / 256, 256, 0, stream>>>(wsf);
    k_borders <<<1,                 128,  0, stream>>>(wsf);
    k_gemm    <<<(MMAX + 15) / 16,  256,  0, stream>>>(feat, wsf, out, newT);
    if (newT > 0)
        k_finalout<<<Bc * newT,     128,  0, stream>>>(wsf, out, newT);
}